// OxDNAEnergy_47966194761957
// MI455X (gfx1250) — compile-verified
//
#include <hip/hip_runtime.h>
#include <math.h>

typedef float v2f __attribute__((ext_vector_type(2)));
typedef float v8f __attribute__((ext_vector_type(8)));

#define PI_F 3.14159265358979323846f
#define EXCL_EPS 2.0f
#define FENE_EPS 2.0f
#define FENE_R0 0.7525f
#define FENE_DELTA 0.25f
#define POS_BACK (-0.4f)
#define POS_STCK 0.34f
#define POS_BASE 0.4f

// f_excl: (sigma, rstar, b, rc)
#define EXCL_BB   0.7f, 0.675f, 892.016223343f, 0.711879214356f
#define EXCL_BASE 0.33f, 0.32f, 4119.70450017f, 0.335388426126f
#define EXCL_MIX  0.515f, 0.5f, 2047.42812499f, 0.52329943261f
// f1/f2: (a|k, r0, rc, rlow, rhigh, blow, rclow, bhigh, rchigh)
#define STCK_F1 6.0f, 0.4f, 0.9f, 0.32f, 0.75f, -0.68f, 0.26f, -12.6f, 0.8f
#define HYDR_F1 8.0f, 0.4f, 0.75f, 0.34f, 0.7f, -126.2f, 0.276f, -7.87f, 0.783f
#define CRST_F2 47.5f, 0.575f, 0.675f, 0.495f, 0.655f, -0.888f, 0.45f, -0.888f, 0.68f
#define CXST_F2 46.0f, 0.4f, 0.6f, 0.22f, 0.58f, -0.7f, 0.2f, -0.7f, 0.62f
// f4: (a, t0, ts, b, tc)
#define F4_STCK_T4 1.3f, 0.0f, 0.8f, 6.4f, 0.961538f
#define F4_STCK_T5 0.9f, 0.0f, 0.95f, 3.9f, 1.16959f
#define F4_HYDR_T1 1.5f, 0.0f, 0.7f, 4.16038f, 0.952381f
#define F4_HYDR_T4 0.46f, PI_F, 0.7f, 1.14813f, 3.0f
#define F4_HYDR_T7 4.0f, 1.57079632679489662f, 0.45f, 17.0526f, 0.555556f
#define F4_CRST_T1 2.25f, 0.791592653589793f, 0.58f, 10.9032f, 0.766284f
#define F4_CRST_T4 1.5f, 0.0f, 0.7f, 4.16038f, 0.952381f
#define F4_CXST_T1 2.0f, 2.592f, 0.65f, 10.9032f, 0.766284f
#define F4_CXST_T4 1.3f, 0.0f, 0.8f, 6.4f, 0.961538f
// f5: (a, xs, b, xc)
#define F5_PHI 2.0f, -0.65f, 10.9032f, -0.769231f

struct V3 { float x, y, z; };
__device__ __forceinline__ V3 mkv3(float x, float y, float z) { V3 r; r.x=x; r.y=y; r.z=z; return r; }
__device__ __forceinline__ V3 vadd(V3 a, V3 b) { return mkv3(a.x+b.x, a.y+b.y, a.z+b.z); }
__device__ __forceinline__ V3 vsub(V3 a, V3 b) { return mkv3(a.x-b.x, a.y-b.y, a.z-b.z); }
__device__ __forceinline__ V3 vscale(float s, V3 a) { return mkv3(s*a.x, s*a.y, s*a.z); }
__device__ __forceinline__ float vdot(V3 a, V3 b) { return a.x*b.x + a.y*b.y + a.z*b.z; }

__device__ __forceinline__ V3 mimg(V3 d, V3 b) {
    d.x -= b.x * rintf(d.x / b.x);
    d.y -= b.y * rintf(d.y / b.y);
    d.z -= b.z * rintf(d.z / b.z);
    return d;
}
__device__ __forceinline__ float norm3(V3 v) { return sqrtf(vdot(v, v) + 1e-12f); }
__device__ __forceinline__ float acosc(float c) {
    return acosf(fminf(fmaxf(c, -1.0f + 1e-6f), 1.0f - 1e-6f));
}

__device__ __forceinline__ float fexcl(float r, float sigma, float rstar, float b, float rc) {
    if (r < rstar) {
        float s2 = (sigma * sigma) / (r * r);
        float s6 = s2 * s2 * s2;
        return 4.0f * EXCL_EPS * (s6 * s6 - s6);
    }
    if (r < rc) { float d = r - rc; return EXCL_EPS * b * d * d; }
    return 0.0f;
}

__device__ __forceinline__ float f1v(float r, float a, float r0, float rc, float rlow,
                                     float rhigh, float blow, float rclow, float bhigh, float rchigh) {
    if (r > rlow && r < rhigh) {
        float sh = expf(-a * (rc - r0)) - 1.0f; sh *= sh;
        float e = expf(-a * (r - r0)) - 1.0f;
        return e * e - sh;
    }
    if (r > rclow && r <= rlow) { float d = r - rclow; return blow * d * d; }
    if (r >= rhigh && r < rchigh) { float d = r - rchigh; return bhigh * d * d; }
    return 0.0f;
}

__device__ __forceinline__ float f2v(float r, float k, float r0, float rc, float rlow,
                                     float rhigh, float blow, float rclow, float bhigh, float rchigh) {
    if (r > rlow && r < rhigh) {
        float d = r - r0, dc = rc - r0;
        return 0.5f * k * (d * d - dc * dc);
    }
    if (r > rclow && r <= rlow) { float d = r - rclow; return k * blow * d * d; }
    if (r >= rhigh && r < rchigh) { float d = r - rchigh; return k * bhigh * d * d; }
    return 0.0f;
}

__device__ __forceinline__ float f4v(float th, float a, float t0, float ts, float b, float tc) {
    float dt = fabsf(th - t0);
    if (dt < ts) return 1.0f - a * dt * dt;
    if (dt < tc) { float d = tc - dt; return b * d * d; }
    return 0.0f;
}

__device__ __forceinline__ float f5v(float x, float a, float xs, float b, float xc) {
    if (x > 0.0f) return 1.0f;
    if (x > xs) return 1.0f - a * x * x;
    if (x > xc) { float d = xc - x; return b * d * d; }
    return 0.0f;
}

__device__ __forceinline__ float block_reduce_256(float v, float* sdata) {
    for (int off = 16; off > 0; off >>= 1) v += __shfl_down(v, off, 32);
    int lane = threadIdx.x & 31;
    int wid = threadIdx.x >> 5;
    if (lane == 0) sdata[wid] = v;
    __syncthreads();
    if (wid == 0) {
        v = (lane < 8) ? sdata[lane] : 0.0f;
        for (int off = 16; off > 0; off >>= 1) v += __shfl_down(v, off, 32);
    }
    return v;
}

// ---------------------------------------------------------------------------
// Kernel 1: per-particle frame precompute -> 64B/particle table in d_ws (L2-resident)
// ---------------------------------------------------------------------------
__global__ void __launch_bounds__(256)
oxdna_prep(const float* __restrict__ pos, const float* __restrict__ quat,
           const int* __restrict__ types, float4* __restrict__ P, int n) {
    int i = blockIdx.x * 256 + threadIdx.x;
    if (i >= n) return;
    float w = quat[4*i+0], x = quat[4*i+1], y = quat[4*i+2], z = quat[4*i+3];
    float inv = 1.0f / sqrtf(w*w + x*x + y*y + z*z + 1e-12f);
    w *= inv; x *= inv; y *= inv; z *= inv;
    float a1x = 1.0f - 2.0f*(y*y + z*z), a1y = 2.0f*(x*y + w*z), a1z = 2.0f*(x*z - w*y);
    float a2x = 2.0f*(x*y - w*z), a2y = 1.0f - 2.0f*(x*x + z*z), a2z = 2.0f*(y*z + w*x);
    float a3x = 2.0f*(x*z + w*y), a3y = 2.0f*(y*z - w*x), a3z = 1.0f - 2.0f*(x*x + y*y);
    size_t b = (size_t)4 * (size_t)i;
    P[b+0] = make_float4(pos[3*i+0], pos[3*i+1], pos[3*i+2], __int_as_float(types[i]));
    P[b+1] = make_float4(a1x, a1y, a1z, 0.0f);
    P[b+2] = make_float4(a2x, a2y, a2z, 0.0f);
    P[b+3] = make_float4(a3x, a3y, a3z, 0.0f);
}

// ---------------------------------------------------------------------------
// Kernel 2: bonded pairs — FENE + excl + stacking
// ---------------------------------------------------------------------------
__global__ void __launch_bounds__(256)
oxdna_bonded(const float4* __restrict__ P, const int* __restrict__ pairs,
             const float* __restrict__ seps, const float* __restrict__ boxp,
             float* __restrict__ partial, int npairs) {
    __shared__ float sdata[8];
    int p = blockIdx.x * 256 + threadIdx.x;
    float e = 0.0f;
    if (p < npairs) {
        int i = pairs[2*p], j = pairs[2*p+1];
        size_t bi = (size_t)4*i, bj = (size_t)4*j;
        float4 i0 = P[bi+0], i1 = P[bi+1], i2 = P[bi+2], i3 = P[bi+3];
        float4 j0 = P[bj+0], j1 = P[bj+1], j2 = P[bj+2], j3 = P[bj+3];
        V3 box = mkv3(boxp[0], boxp[1], boxp[2]);
        V3 dp = mkv3(j0.x-i0.x, j0.y-i0.y, j0.z-i0.z);
        V3 a1i = mkv3(i1.x,i1.y,i1.z), a2i = mkv3(i2.x,i2.y,i2.z), a3i = mkv3(i3.x,i3.y,i3.z);
        V3 a1j = mkv3(j1.x,j1.y,j1.z), a2j = mkv3(j2.x,j2.y,j2.z), a3j = mkv3(j3.x,j3.y,j3.z);
        V3 da1 = vsub(a1j, a1i), sa1 = vadd(a1j, a1i);

        // FENE on backbone sites
        V3 dback = mimg(vadd(dp, vscale(POS_BACK, da1)), box);
        float rbb = norm3(dback);
        float xx = (rbb - FENE_R0) / FENE_DELTA;
        float arg = fminf(fmaxf(xx * xx, 0.0f), 1.0f - 1e-6f);
        e += -0.5f * FENE_EPS * log1pf(-arg);

        // bonded excluded volume
        e += fexcl(norm3(mimg(vadd(dp, vscale(POS_BASE, da1)), box)), EXCL_BASE); // base-base
        e += fexcl(norm3(mimg(vadd(dp, vscale(POS_BASE, sa1)), box)), EXCL_MIX);  // base_j - back_i
        e += fexcl(norm3(mimg(vsub(dp, vscale(POS_BASE, sa1)), box)), EXCL_MIX);  // back_j - base_i

        // stacking
        V3 ds = mimg(vadd(dp, vscale(POS_STCK, da1)), box);
        float r = norm3(ds);
        V3 rhat = vscale(1.0f / r, ds);
        float t4 = acosc(vdot(a3i, a3j));
        float t5 = acosc(vdot(a3j, rhat));
        float t6 = acosc(-vdot(a3i, rhat));
        V3 rbhat = vscale(1.0f / rbb, dback);
        float cphi1 = vdot(a2i, rbhat), cphi2 = vdot(a2j, rbhat);
        e += seps[p] * f1v(r, STCK_F1) * f4v(t4, F4_STCK_T4) * f4v(t5, F4_STCK_T5)
                     * f4v(t6, F4_STCK_T5) * f5v(cphi1, F5_PHI) * f5v(cphi2, F5_PHI);
    }
    float bsum = block_reduce_256(e, sdata);
    if (threadIdx.x == 0) partial[blockIdx.x] = bsum;
}

// ---------------------------------------------------------------------------
// Kernel 3: nonbonded pairs — excl + H-bond + cross-stack + coaxial
// ---------------------------------------------------------------------------
__global__ void __launch_bounds__(256)
oxdna_nonbonded(const float4* __restrict__ P, const int* __restrict__ pairs,
                const float* __restrict__ hbe, const float* __restrict__ boxp,
                float* __restrict__ partial, int npairs) {
    __shared__ float sdata[8];
    int p = blockIdx.x * 256 + threadIdx.x;
    float e = 0.0f;
    if (p < npairs) {
        int i = pairs[2*p], j = pairs[2*p+1];
        size_t bi = (size_t)4*i, bj = (size_t)4*j;
        float4 i0 = P[bi+0], i1 = P[bi+1], i2 = P[bi+2], i3 = P[bi+3];
        float4 j0 = P[bj+0], j1 = P[bj+1], j2 = P[bj+2], j3 = P[bj+3];
        V3 box = mkv3(boxp[0], boxp[1], boxp[2]);
        V3 dp = mkv3(j0.x-i0.x, j0.y-i0.y, j0.z-i0.z);
        V3 a1i = mkv3(i1.x,i1.y,i1.z), a2i = mkv3(i2.x,i2.y,i2.z), a3i = mkv3(i3.x,i3.y,i3.z);
        V3 a1j = mkv3(j1.x,j1.y,j1.z), a2j = mkv3(j2.x,j2.y,j2.z), a3j = mkv3(j3.x,j3.y,j3.z);
        V3 da1 = vsub(a1j, a1i), sa1 = vadd(a1j, a1i);

        // nonbonded excluded volume (back-back, base-base, mixed x2)
        e += fexcl(norm3(mimg(vadd(dp, vscale(POS_BACK, da1)), box)), EXCL_BB);
        V3 dbase = mimg(vadd(dp, vscale(POS_BASE, da1)), box);
        float r = norm3(dbase);
        e += fexcl(r, EXCL_BASE);
        e += fexcl(norm3(mimg(vadd(dp, vscale(POS_BASE, sa1)), box)), EXCL_MIX);
        e += fexcl(norm3(mimg(vsub(dp, vscale(POS_BASE, sa1)), box)), EXCL_MIX);

        // hydrogen bond + cross-stacking angles (base-base axis)
        V3 rhat = vscale(1.0f / r, dbase);
        float t1 = acosc(-vdot(a1i, a1j));
        float t2 = acosc(-vdot(a1j, rhat));
        float t3 = acosc(vdot(a1i, rhat));
        float t4 = acosc(vdot(a3i, a3j));
        float t7 = acosc(-vdot(a3j, rhat));
        float t8 = acosc(vdot(a3i, rhat));
        int ti = __float_as_int(i0.w), tj = __float_as_int(j0.w);
        float eps_hb = hbe[ti * 4 + tj];
        float f4t7 = f4v(t7, F4_HYDR_T7);
        e += eps_hb * f1v(r, HYDR_F1) * f4v(t1, F4_HYDR_T1) * f4v(t2, F4_HYDR_T1)
                    * f4v(t3, F4_HYDR_T1) * f4v(t4, F4_HYDR_T4) * f4t7 * f4v(t8, F4_HYDR_T7);
        e += f2v(r, CRST_F2) * f4v(t1, F4_CRST_T1) * f4v(t4, F4_CRST_T4)
             * (f4t7 + f4v(PI_F - t7, F4_HYDR_T7));

        // coaxial stacking (stack-site axis); ct1 == t1, ct4 == t4
        V3 dcx = mimg(vadd(dp, vscale(POS_STCK, da1)), box);
        float rcx = norm3(dcx);
        V3 rchat = vscale(1.0f / rcx, dcx);
        float ct5 = acosc(vdot(a3j, rchat));
        float cphi3 = vdot(a2i, a2j);
        e += f2v(rcx, CXST_F2) * f4v(t1, F4_CXST_T1) * f4v(t4, F4_CXST_T4)
             * f4v(ct5, F4_STCK_T5) * f5v(cphi3, F5_PHI);
    }
    float bsum = block_reduce_256(e, sdata);
    if (threadIdx.x == 0) partial[blockIdx.x] = bsum;
}

// ---------------------------------------------------------------------------
// Kernel 4: deterministic final reduction; 64 per-lane accumulators collapsed
// with V_WMMA_F32_16X16X4_F32 against a ones B-matrix (D[m][n] = sum_k A[m][k]).
// ---------------------------------------------------------------------------
__global__ void oxdna_finalreduce(const float* __restrict__ partial, int n, float* __restrict__ out) {
    int lane = threadIdx.x; // exactly 32 threads, EXEC all-ones at the WMMA
    float s0 = 0.0f, s1 = 0.0f;
    for (int i = lane; i < n; i += 64) s0 += partial[i];
    for (int i = lane + 32; i < n; i += 64) s1 += partial[i];
#if defined(__gfx1250__)
    v2f a; a[0] = s0; a[1] = s1;          // A: 16x4 f32 striped over 32 lanes (2 VGPRs)
    v2f b; b[0] = 1.0f; b[1] = 1.0f;      // B: 4x16 all-ones
    v8f c = {};
    c = __builtin_amdgcn_wmma_f32_16x16x4_f32(false, a, false, b, (short)0, c, false, false);
    // Row sums replicated across columns: lanes 0-15 hold rows 0-7 in c[0..7],
    // lanes 16-31 hold rows 8-15. Sum this lane's 8 rows, add the other half.
    float t = c[0] + c[1] + c[2] + c[3] + c[4] + c[5] + c[6] + c[7];
    float u = __shfl(t, 16, 32);
    if (lane == 0) out[0] = t + u;
#else
    float t = s0 + s1;
    for (int off = 16; off > 0; off >>= 1) t += __shfl_down(t, off, 32);
    if (lane == 0) out[0] = t;
#endif
}

extern "C" void kernel_launch(void* const* d_in, const int* in_sizes, int n_in,
                              void* d_out, int out_size, void* d_ws, size_t ws_size,
                              hipStream_t stream) {
    const float* pos   = (const float*)d_in[0];
    const float* quat  = (const float*)d_in[1];
    const float* seps  = (const float*)d_in[2];
    const float* hbe   = (const float*)d_in[3];
    const float* box   = (const float*)d_in[4];
    const int* bpairs  = (const int*)d_in[5];
    const int* nbpairs = (const int*)d_in[6];
    const int* types   = (const int*)d_in[7];

    int n   = in_sizes[7];      // particles
    int nb  = in_sizes[5] / 2;  // bonded pairs
    int nnb = in_sizes[6] / 2;  // nonbonded pairs

    float4* P = (float4*)d_ws;                                     // n * 64 B
    float* partial = (float*)((char*)d_ws + (size_t)n * 64);       // gridB + gridN floats

    int gridP = (n + 255) / 256;
    int gridB = (nb + 255) / 256;
    int gridN = (nnb + 255) / 256;

    oxdna_prep<<<gridP, 256, 0, stream>>>(pos, quat, types, P, n);
    oxdna_bonded<<<gridB, 256, 0, stream>>>(P, bpairs, seps, box, partial, nb);
    oxdna_nonbonded<<<gridN, 256, 0, stream>>>(P, nbpairs, hbe, box, partial + gridB, nnb);
    oxdna_finalreduce<<<1, 32, 0, stream>>>(partial, gridB + gridN, (float*)d_out);
}